// GCN_SP_three_86887188398704
// MI455X (gfx1250) — compile-verified
//
#include <hip/hip_runtime.h>
#include <math.h>

typedef __attribute__((ext_vector_type(2))) float v2f;
typedef __attribute__((ext_vector_type(8))) float v8f;

#define N_NODES 50000
#define N_EDGES 800000

// Max padded LDS row stride: K=500 -> S=516
#define MAX_LDS_STRIDE 516

// ---------------------------------------------------------------------------
// Zero-fill, float4-wide (all buffer sizes are multiples of 4)
// ---------------------------------------------------------------------------
__global__ void zero_f32v4(float4* __restrict__ p, long long n4) {
  long long i = (long long)blockIdx.x * blockDim.x + threadIdx.x;
  if (i < n4) p[i] = make_float4(0.f, 0.f, 0.f, 0.f);
}

// ---------------------------------------------------------------------------
// D[M,N] = A[M,K] @ B[K,N]  (row-major fp32) via V_WMMA_F32_16X16X4_F32.
//
// Each block stages its 16 A-rows into LDS with coalesced float4 global
// reads (A read exactly once from HBM), then each wave32 computes one 16x16
// output tile, fetching A-fragments with conflict-free ds_load_b64.
//
// LDS row stride S = round_up(K,64)+4  =>  S % 64 == 4, so lane r's b64 pair
// hits banks {4r+2hi, 4r+2hi+1}: all 64 banks, no conflicts, for every K.
//
// A-fragment (16x4): lanes 0-15 carry K={k,k+1}, lanes 16-31 K={k+2,k+3}.
// B-fragment (4x16): lane holds column nBase+(lane&15), rows k+2hi, k+2hi+1.
// C/D (16x16 f32): VGPR v = row v (lanes 0-15) / row v+8 (lanes 16-31).
// Requires M%16==0, K%4==0, N%16==0 (true for all shapes here).
// ---------------------------------------------------------------------------
__global__ void gemm_wmma_f32(const float* __restrict__ A, const float* __restrict__ B,
                              float* __restrict__ D, int M, int K, int N) {
  __shared__ float ldsA[16 * MAX_LDS_STRIDE];
  const int S = ((K + 63) & ~63) + 4;      // padded row stride (S % 64 == 4)
  const int mBase = blockIdx.x << 4;

  // ---- cooperative, coalesced stage of A[mBase:mBase+16, :] into LDS ----
  {
    const int K4 = K >> 2;
    for (int row = 0; row < 16; ++row) {
      const float4* src = (const float4*)(A + (size_t)(mBase + row) * K);
      float4* dst = (float4*)(ldsA + row * S);
      for (int c4 = threadIdx.x; c4 < K4; c4 += blockDim.x) dst[c4] = src[c4];
    }
  }
  __syncthreads();

  const int wavesPerBlock = blockDim.x >> 5;
  const int wave = threadIdx.x >> 5;
  const int lane = threadIdx.x & 31;
  const int nBase = (blockIdx.y * wavesPerBlock + wave) << 4;
  if (nBase >= N) return;                  // wave-uniform: EXEC stays all-ones

  const int r  = lane & 15;
  const int hi = lane >> 4;                // half-wave K sub-pair selector

  const float* Afrag = ldsA + r * S + (hi << 1);            // LDS (ds_load_b64)
  const float* Bcol  = B + nBase + r + (size_t)(hi << 1) * N;

  v8f c = {};
  for (int k = 0; k < K; k += 4) {
    v2f a = *(const v2f*)(Afrag + k);      // ds_load_b64, bank-conflict-free
    v2f b;
    b.x = Bcol[(size_t)k * N];             // B[k+2*hi  ][nBase+r]  (L2-resident)
    b.y = Bcol[(size_t)k * N + N];         // B[k+2*hi+1][nBase+r]
    c = __builtin_amdgcn_wmma_f32_16x16x4_f32(false, a, false, b, (short)0, c,
                                              false, false);
  }

  float* Dp = D + (size_t)(mBase + (hi << 3)) * N + nBase + r;
#pragma unroll
  for (int v = 0; v < 8; ++v) Dp[(size_t)v * N] = c[v];
}

// ---------------------------------------------------------------------------
// agg[src[e], :] += support[dst[e], :] * val[e]
// One thread per (edge, 4-wide feature chunk). Accumulators are L2-resident
// (192 MB L2 >> 83 MB working set) so the f32 atomics resolve in L2.
// ---------------------------------------------------------------------------
__global__ void spmm_scatter(const float* __restrict__ S, const int* __restrict__ src,
                             const int* __restrict__ dst, const float* __restrict__ vals,
                             float* __restrict__ agg, int E, int d) {
  const int chunks = d >> 2;
  long long tid = (long long)blockIdx.x * blockDim.x + threadIdx.x;
  long long total = (long long)E * chunks;
  if (tid >= total) return;
  int e = (int)(tid / chunks);
  int c = (int)(tid - (long long)e * chunks) << 2;
  float v = vals[e];
  int sN = src[e];
  int dN = dst[e];
  const float4 m = *(const float4*)(S + (size_t)dN * d + c);
  float* ap = agg + (size_t)sN * d + c;
  atomicAdd(ap + 0, m.x * v);
  atomicAdd(ap + 1, m.y * v);
  atomicAdd(ap + 2, m.z * v);
  atomicAdd(ap + 3, m.w * v);
}

// ---------------------------------------------------------------------------
// X[i4] = act(X[i4] + bias[(i4*4) % d ...]); float4-wide; d % 4 == 0
// ---------------------------------------------------------------------------
__global__ void bias_act4(float4* __restrict__ X, const float* __restrict__ bias,
                          long long total4, int d4, int relu) {
  long long i = (long long)blockIdx.x * blockDim.x + threadIdx.x;
  if (i >= total4) return;
  int c = (int)(i % d4) << 2;
  float4 v = X[i];
  v.x += bias[c + 0];
  v.y += bias[c + 1];
  v.z += bias[c + 2];
  v.w += bias[c + 3];
  if (relu) {
    v.x = fmaxf(v.x, 0.f); v.y = fmaxf(v.y, 0.f);
    v.z = fmaxf(v.z, 0.f); v.w = fmaxf(v.w, 0.f);
  }
  X[i] = v;
}

// ---------------------------------------------------------------------------
// out[row,:] = log_softmax(agg[row,:] + b3)  for 16-wide rows; 1 thread / row
// ---------------------------------------------------------------------------
__global__ void bias_logsoftmax16(const float* __restrict__ agg, const float* __restrict__ b,
                                  float* __restrict__ out, int M) {
  int row = blockIdx.x * blockDim.x + threadIdx.x;
  if (row >= M) return;
  float v[16];
  float mx = -INFINITY;
#pragma unroll
  for (int j = 0; j < 16; ++j) {
    v[j] = agg[(size_t)row * 16 + j] + b[j];
    mx = fmaxf(mx, v[j]);
  }
  float s = 0.0f;
#pragma unroll
  for (int j = 0; j < 16; ++j) s += expf(v[j] - mx);
  float ls = logf(s);
#pragma unroll
  for (int j = 0; j < 16; ++j) out[(size_t)row * 16 + j] = v[j] - mx - ls;
}

// ---------------------------------------------------------------------------
static inline void launch_gemm(const float* A, const float* B, float* D,
                               int M, int K, int N, hipStream_t s) {
  int nTiles = N >> 4;
  int wavesN = nTiles < 8 ? nTiles : 8;
  dim3 block(32 * wavesN);
  dim3 grid(M >> 4, (nTiles + wavesN - 1) / wavesN);
  gemm_wmma_f32<<<grid, block, 0, s>>>(A, B, D, M, K, N);
}

static inline void launch_zero(float* p, long long n, hipStream_t s) {
  long long n4 = n >> 2;
  zero_f32v4<<<(int)((n4 + 255) / 256), 256, 0, s>>>((float4*)p, n4);
}

static inline void launch_spmm(const float* S, const int* src, const int* dst,
                               const float* vals, float* agg, int E, int d,
                               hipStream_t s) {
  long long total = (long long)E * (d >> 2);
  spmm_scatter<<<(int)((total + 255) / 256), 256, 0, s>>>(S, src, dst, vals, agg, E, d);
}

static inline void launch_bias_act(float* X, const float* bias, long long total,
                                   int d, int relu, hipStream_t s) {
  long long total4 = total >> 2;
  bias_act4<<<(int)((total4 + 255) / 256), 256, 0, s>>>((float4*)X, bias, total4,
                                                        d >> 2, relu);
}

extern "C" void kernel_launch(void* const* d_in, const int* in_sizes, int n_in,
                              void* d_out, int out_size, void* d_ws, size_t ws_size,
                              hipStream_t stream) {
  const float* x    = (const float*)d_in[0];   // [50000, 500]
  const float* ev   = (const float*)d_in[1];   // [800000]
  const float* W1   = (const float*)d_in[2];   // [500, 128]
  const float* b1   = (const float*)d_in[3];   // [128]
  const float* W2   = (const float*)d_in[4];   // [128, 64]
  const float* b2   = (const float*)d_in[5];   // [64]
  const float* W3   = (const float*)d_in[6];   // [64, 16]
  const float* b3   = (const float*)d_in[7];   // [16]
  const float* We   = (const float*)d_in[8];   // [64, 64]
  const float* be   = (const float*)d_in[9];   // [64]
  const int*   esrc = (const int*)d_in[10];    // [800000]
  const int*   edst = (const int*)d_in[11];    // [800000]

  const int M = N_NODES, E = N_EDGES;

  // Workspace layout (floats)
  float* ws   = (float*)d_ws;
  float* s1   = ws;                          // [M,128]
  float* h1   = s1 + (size_t)M * 128;        // [M,128]  agg1 -> relu in place
  float* s2   = h1 + (size_t)M * 128;        // [M,64]
  float* h2   = s2 + (size_t)M * 64;         // [M,64]
  float* s3   = h2 + (size_t)M * 64;         // [M,16]
  float* agg3 = s3 + (size_t)M * 16;         // [M,16]

  float* out1 = (float*)d_out;               // [M,16]
  float* out2 = out1 + (size_t)M * 16;       // [M,64]

  // ---- Layer 1: relu(spmm(x@W1) + b1) ----
  launch_gemm(x, W1, s1, M, 500, 128, stream);
  launch_zero(h1, (long long)M * 128, stream);
  launch_spmm(s1, esrc, edst, ev, h1, E, 128, stream);
  launch_bias_act(h1, b1, (long long)M * 128, 128, 1, stream);

  // ---- Layer 2: relu(spmm(h1@W2) + b2) ----
  launch_gemm(h1, W2, s2, M, 128, 64, stream);
  launch_zero(h2, (long long)M * 64, stream);
  launch_spmm(s2, esrc, edst, ev, h2, E, 64, stream);
  launch_bias_act(h2, b2, (long long)M * 64, 64, 1, stream);

  // ---- Layer 3: log_softmax(spmm(h2@W3) + b3) -> out1 ----
  launch_gemm(h2, W3, s3, M, 64, 16, stream);
  launch_zero(agg3, (long long)M * 16, stream);
  launch_spmm(s3, esrc, edst, ev, agg3, E, 16, stream);
  bias_logsoftmax16<<<(M + 255) / 256, 256, 0, stream>>>(agg3, b3, out1, M);

  // ---- Head: out2 = h2 @ We + be ----
  launch_gemm(h2, We, out2, M, 64, 64, stream);
  launch_bias_act(out2, be, (long long)M * 64, 64, 0, stream);
}